// Sensory2Hippo_87436944212024
// MI455X (gfx1250) — compile-verified
//
#include <hip/hip_runtime.h>

// Problem constants (from reference)
#define BATCH   2048
#define XF_DIM  1024
#define GTILE   16      // output = 16x feature tile
#define OUT_DIM (XF_DIM * GTILE)

typedef __attribute__((ext_vector_type(2))) float v2f;
typedef __attribute__((ext_vector_type(8))) float v8f;

// ---------------------------------------------------------------------------
// Pass 1: column means of xf via V_WMMA_F32_16X16X4_F32 with A = ones(16x4).
// D[m][n] = sum_k B[k][n]  (all D rows identical). Each wave owns a 16-col
// tile; 8 waves/block split the 2048 rows; LDS reduce across waves.
// grid = 64 blocks (1024/16 cols), block = 256 threads (8 waves).
// ---------------------------------------------------------------------------
__global__ void __launch_bounds__(256)
colmean_wmma_kernel(const float* __restrict__ xf, float* __restrict__ mean)
{
    const int lane  = threadIdx.x & 31;
    const int wave  = threadIdx.x >> 5;
    const int half  = lane >> 4;          // 0: lanes 0-15, 1: lanes 16-31
    const int cbase = blockIdx.x * 16;
    const int col   = cbase + (lane & 15);

    const v2f ones = {1.0f, 1.0f};
    v8f acc = {};

    // Each wave reduces 256 rows, 4 rows per WMMA (K=4).
    const int r0 = wave * (BATCH / 8);
    const int r1 = r0 + (BATCH / 8);
    for (int r = r0; r < r1; r += 4) {
        // B is 4x16: lane (n = lane&15) holds two K elements; any K<->VGPR
        // bijection is valid since A is all-ones (we sum over K).
        v2f b;
        b.x = xf[(r + 2 * half + 0) * XF_DIM + col];
        b.y = xf[(r + 2 * half + 1) * XF_DIM + col];
        acc = __builtin_amdgcn_wmma_f32_16x16x4_f32(
            false, ones, false, b, (short)0, acc, false, false);
    }

    // acc[0] at lane L = partial column sum for column cbase + (L&15).
    __shared__ float part[8][16];
    if (lane < 16) part[wave][lane] = acc[0];
    __syncthreads();

    if (threadIdx.x < 16) {
        float s = 0.0f;
        #pragma unroll
        for (int w = 0; w < 8; ++w) s += part[w][threadIdx.x];
        mean[cbase + threadIdx.x] = s * (1.0f / (float)BATCH);
    }
}

// ---------------------------------------------------------------------------
// Pass 2: per-column sum and sum-of-squares of v = relu(x - colmean), again
// via all-ones WMMA (two accumulators). Then mu, unbiased std (ddof=1),
// emit per-column scale A = wp/sd and bias B = -A*mu.
// grid = 64, block = 256.
// ---------------------------------------------------------------------------
__global__ void __launch_bounds__(256)
colstats_wmma_kernel(const float* __restrict__ xf,
                     const float* __restrict__ mean,
                     const float* __restrict__ wp,
                     float* __restrict__ scaleA,
                     float* __restrict__ biasB)
{
    const int lane  = threadIdx.x & 31;
    const int wave  = threadIdx.x >> 5;
    const int half  = lane >> 4;
    const int cbase = blockIdx.x * 16;
    const int col   = cbase + (lane & 15);

    const float m = mean[col];
    const v2f ones = {1.0f, 1.0f};
    v8f acc_s = {};
    v8f acc_q = {};

    const int r0 = wave * (BATCH / 8);
    const int r1 = r0 + (BATCH / 8);
    for (int r = r0; r < r1; r += 4) {
        float v0 = fmaxf(xf[(r + 2 * half + 0) * XF_DIM + col] - m, 0.0f);
        float v1 = fmaxf(xf[(r + 2 * half + 1) * XF_DIM + col] - m, 0.0f);
        v2f b  = {v0, v1};
        v2f bq = {v0 * v0, v1 * v1};
        acc_s = __builtin_amdgcn_wmma_f32_16x16x4_f32(
            false, ones, false, b, (short)0, acc_s, false, false);
        acc_q = __builtin_amdgcn_wmma_f32_16x16x4_f32(
            false, ones, false, bq, (short)0, acc_q, false, false);
    }

    __shared__ float ps[8][16];
    __shared__ float pq[8][16];
    if (lane < 16) {
        ps[wave][lane] = acc_s[0];
        pq[wave][lane] = acc_q[0];
    }
    __syncthreads();

    if (threadIdx.x < 16) {
        float s = 0.0f, q = 0.0f;
        #pragma unroll
        for (int w = 0; w < 8; ++w) { s += ps[w][threadIdx.x]; q += pq[w][threadIdx.x]; }
        const float mu  = s * (1.0f / (float)BATCH);
        // unbiased variance: (sum(v^2) - N*mu^2) / (N-1)
        const float var = (q - (float)BATCH * mu * mu) * (1.0f / (float)(BATCH - 1));
        const float a   = wp[0] * rsqrtf(var);
        const int c = cbase + threadIdx.x;
        scaleA[c] = a;
        biasB[c]  = -a * mu;
    }
}

// ---------------------------------------------------------------------------
// Pass 3 (bandwidth-dominant): y = A*relu(x - mean) + B, replicated 16x along
// the feature axis. float4 per thread: 256 threads cover one 1024-col row;
// each wave store instruction = 32 lanes x 16B = 512B contiguous.
// grid = 2048 (one block per row), block = 256.
// ---------------------------------------------------------------------------
__global__ void __launch_bounds__(256)
tile_write_kernel(const float* __restrict__ xf,
                  const float* __restrict__ mean,
                  const float* __restrict__ scaleA,
                  const float* __restrict__ biasB,
                  float* __restrict__ out)
{
    const int row = blockIdx.x;
    const int c4  = threadIdx.x * 4;

    const float4 x = *(const float4*)(xf + row * XF_DIM + c4);
    const float4 M = *(const float4*)(mean + c4);
    const float4 A = *(const float4*)(scaleA + c4);
    const float4 B = *(const float4*)(biasB + c4);

    float4 y;
    y.x = fmaf(fmaxf(x.x - M.x, 0.0f), A.x, B.x);
    y.y = fmaf(fmaxf(x.y - M.y, 0.0f), A.y, B.y);
    y.z = fmaf(fmaxf(x.z - M.z, 0.0f), A.z, B.z);
    y.w = fmaf(fmaxf(x.w - M.w, 0.0f), A.w, B.w);

    float4* dst = (float4*)(out + (size_t)row * OUT_DIM + c4);
    #pragma unroll
    for (int g = 0; g < GTILE; ++g) {
        dst[g * (XF_DIM / 4)] = y;
    }
}

extern "C" void kernel_launch(void* const* d_in, const int* in_sizes, int n_in,
                              void* d_out, int out_size, void* d_ws, size_t ws_size,
                              hipStream_t stream)
{
    (void)in_sizes; (void)n_in; (void)out_size; (void)ws_size;

    const float* xf = (const float*)d_in[0];   // [2048, 1024] f32
    const float* wp = (const float*)d_in[1];   // [1] f32
    float* out = (float*)d_out;                // [2048, 16384] f32

    float* ws     = (float*)d_ws;
    float* mean   = ws;             // [1024]
    float* scaleA = ws + XF_DIM;    // [1024]
    float* biasB  = ws + 2 * XF_DIM;// [1024]

    colmean_wmma_kernel<<<XF_DIM / 16, 256, 0, stream>>>(xf, mean);
    colstats_wmma_kernel<<<XF_DIM / 16, 256, 0, stream>>>(xf, mean, wp, scaleA, biasB);
    tile_write_kernel<<<BATCH, 256, 0, stream>>>(xf, mean, scaleA, biasB, out);
}